// JointBilateralFilter_50388556316755
// MI455X (gfx1250) — compile-verified
//
#include <hip/hip_runtime.h>
#include <math.h>

// Joint bilateral filter: B=16, C=1, H=768, W=1024, 9x9 window.
// sigma_s=2 -> spatial exponent -(i^2+j^2)/8 ; sigma_r=0.1 -> range coef 50.
// One v_exp_f32 per tap: w = exp2( fma(diff^2, -50*log2e, -(i^2+j^2)*log2e/8) ).

constexpr int BATCH  = 16;
constexpr int HH     = 768;
constexpr int WW     = 1024;
constexpr int PADR   = 4;
constexpr int TILE_W = 64;            // output tile width  (16 threads x 4 px)
constexpr int TILE_H = 16;            // output tile height (16 threads)
constexpr int LDS_W  = TILE_W + 2 * PADR;   // 72
constexpr int LDS_H  = TILE_H + 2 * PADR;   // 24

// -50 * log2(e), and log2(e)/8
#define RANGE_C2   (-72.134752044448169f)
#define SPATIAL_C2 (0.18033688011112043f)

typedef __attribute__((ext_vector_type(4))) unsigned int v4u;
typedef __attribute__((ext_vector_type(4))) int          v4i;
typedef __attribute__((ext_vector_type(8))) int          v8i;

// Device-pass-only feature gates (host pass must never see amdgcn builtins).
#if defined(__HIP_DEVICE_COMPILE__) && __has_builtin(__builtin_amdgcn_tensor_load_to_lds)
#define HAVE_TDM 1
#else
#define HAVE_TDM 0
#endif

#if defined(__HIP_DEVICE_COMPILE__) && __has_builtin(__builtin_amdgcn_exp2f)
#define HAVE_EXP2 1
#else
#define HAVE_EXP2 0
#endif

__device__ __forceinline__ float fast_exp2(float x) {
#if HAVE_EXP2
  return __builtin_amdgcn_exp2f(x);      // v_exp_f32
#else
  return exp2f(x);
#endif
}

#if HAVE_TDM
// Issue one TDM 2D tile load: LDS_H rows x LDS_W floats, global row stride WW.
// D# per CDNA5 ISA ch.8: group0 = {count|flags, lds_addr, global_addr_lo,
// global_addr_hi|type=2}; group1 packs data_size/tensor dims/tile dims/stride.
__device__ __forceinline__ void tdm_load_tile(const float* gptr, unsigned lds_byte) {
  unsigned long long ga = (unsigned long long)(__SIZE_TYPE__)gptr;
  v4u g0;
  g0.x = 1u;                                         // count=1, user mode
  g0.y = lds_byte;                                   // lds_addr (bytes)
  g0.z = (unsigned)(ga & 0xFFFFFFFFu);               // global_addr[31:0]
  g0.w = (unsigned)((ga >> 32) & 0x01FFFFFFu)        // global_addr[56:32]
       | 0x80000000u;                                // type=2 ("image")
  v8i g1;
  g1.s0 = 0x00020000;                                // data_size=2 (4 bytes)
  g1.s1 = 0;                                         // tensor_dim0 = 1<<20 (lo16<<16)
  g1.s2 = 0x00000010;                                // dim0 hi16 | dim1 lo16<<16
  g1.s3 = 0x00000010 | (LDS_W << 16);                // dim1 hi16 | tile_dim0
  g1.s4 = LDS_H;                                     // tile_dim1 | tile_dim2=0
  g1.s5 = WW;                                        // tensor_dim0_stride lo32
  g1.s6 = 0;                                         // stride hi16 | dim1_stride lo16
  g1.s7 = 0;
  v4i g2 = {0, 0, 0, 0};                             // unused (2D tile)
  v4i g3 = {0, 0, 0, 0};
#if __clang_major__ >= 23
  v8i g4 = {0, 0, 0, 0, 0, 0, 0, 0};
  __builtin_amdgcn_tensor_load_to_lds(g0, g1, g2, g3, g4, 0);
#else
  __builtin_amdgcn_tensor_load_to_lds(g0, g1, g2, g3, 0);
#endif
}
#endif

__global__ __launch_bounds__(256)
void jbf_kernel(const float* __restrict__ sparse,
                const float* __restrict__ depth,
                float* __restrict__ out) {
  __shared__ float lds_sp[LDS_H * LDS_W];
  __shared__ float lds_dp[LDS_H * LDS_W];

  const int bx = blockIdx.x, by = blockIdx.y, bz = blockIdx.z;
  const int tx = threadIdx.x, ty = threadIdx.y;     // 16 x 16 threads
  const int tid = ty * 16 + tx;
  const size_t imgbase = (size_t)bz * (size_t)(HH * WW);
  const int x0 = bx * TILE_W;
  const int y0 = by * TILE_H;

  const bool interior = (x0 >= PADR) && (y0 >= PADR) &&
                        (x0 + TILE_W + PADR <= WW) &&
                        (y0 + TILE_H + PADR <= HH);
  (void)interior;

#if HAVE_TDM
  if (interior) {
    if (tid < 32) {   // wave 0 issues the DMA (TDM ignores EXEC, wave-level op)
      const size_t toff = imgbase + (size_t)(y0 - PADR) * WW + (x0 - PADR);
      tdm_load_tile(sparse + toff, (unsigned)(__SIZE_TYPE__)(void*)lds_sp);
      tdm_load_tile(depth  + toff, (unsigned)(__SIZE_TYPE__)(void*)lds_dp);
#if __has_builtin(__builtin_amdgcn_s_wait_tensorcnt)
      __builtin_amdgcn_s_wait_tensorcnt(0);
#else
      asm volatile("s_wait_tensorcnt 0x0" ::: "memory");
#endif
    }
  } else
#endif
  {
    // Boundary tiles: manual staging. depth uses 'reflect' padding,
    // sparse uses constant 1.0 (== invalid) outside the image.
    for (int i = tid; i < LDS_H * LDS_W; i += 256) {
      const int ly = i / LDS_W;
      const int lx = i - ly * LDS_W;
      const int gy = y0 - PADR + ly;
      const int gx = x0 - PADR + lx;
      int ry = gy < 0 ? -gy : gy;       ry = ry >= HH ? (2 * HH - 2 - ry) : ry;
      int rx = gx < 0 ? -gx : gx;       rx = rx >= WW ? (2 * WW - 2 - rx) : rx;
      lds_dp[i] = depth[imgbase + (size_t)ry * WW + rx];
      const int cy = gy < 0 ? 0 : (gy >= HH ? HH - 1 : gy);
      const int cx = gx < 0 ? 0 : (gx >= WW ? WW - 1 : gx);
      const float sv = sparse[imgbase + (size_t)cy * WW + cx];
      const bool inb = (gy >= 0) && (gy < HH) && (gx >= 0) && (gx < WW);
      lds_sp[i] = inb ? sv : 1.0f;
    }
  }
  __syncthreads();

  // Each thread: 4 consecutive x outputs at row ty, cols 4*tx .. 4*tx+3.
  const int lbase = (ty + PADR) * LDS_W + 4 * tx + PADR;
  const float4 dcv = *(const float4*)&lds_dp[lbase];
  const float dc[4] = {dcv.x, dcv.y, dcv.z, dcv.w};

  float num[4] = {0.f, 0.f, 0.f, 0.f};
  float den[4] = {0.f, 0.f, 0.f, 0.f};

  for (int dy = 0; dy < 9; ++dy) {
    const int rb = (ty + dy) * LDS_W + 4 * tx;      // 16B-aligned
    const float4 s0 = *(const float4*)&lds_sp[rb];
    const float4 s1 = *(const float4*)&lds_sp[rb + 4];
    const float4 s2 = *(const float4*)&lds_sp[rb + 8];
    const float4 p0 = *(const float4*)&lds_dp[rb];
    const float4 p1 = *(const float4*)&lds_dp[rb + 4];
    const float4 p2 = *(const float4*)&lds_dp[rb + 8];
    const float spw[12] = {s0.x, s0.y, s0.z, s0.w, s1.x, s1.y, s1.z, s1.w,
                           s2.x, s2.y, s2.z, s2.w};
    const float dpw[12] = {p0.x, p0.y, p0.z, p0.w, p1.x, p1.y, p1.z, p1.w,
                           p2.x, p2.y, p2.z, p2.w};
    const int dyr = dy - PADR;
#pragma unroll
    for (int dx = 0; dx < 9; ++dx) {
      const int dxr = dx - PADR;
      const float swl = (float)(dyr * dyr + dxr * dxr) * (-SPATIAL_C2);
#pragma unroll
      for (int c = 0; c < 4; ++c) {
        const float s    = spw[dx + c];
        const float d    = dpw[dx + c];
        const float diff = d - dc[c];
        const float e    = __builtin_fmaf(diff * diff, RANGE_C2, swl);
        float w          = fast_exp2(e);          // spatial*range in one exp
        w = (s != 1.0f) ? w : 0.0f;               // validity select
        num[c] = __builtin_fmaf(w, s, num[c]);
        den[c] += w;
      }
    }
  }

  float4 o;
  float* po = (float*)&o;
#pragma unroll
  for (int c = 0; c < 4; ++c) {
    const float d = den[c];
    const float r = num[c] / (d + 1e-8f);
    po[c] = (d < 1e-8f) ? 1.0f : r;
  }
  const int gy = y0 + ty;
  const int gx = x0 + 4 * tx;
  *(float4*)&out[imgbase + (size_t)gy * WW + gx] = o;
}

extern "C" void kernel_launch(void* const* d_in, const int* in_sizes, int n_in,
                              void* d_out, int out_size, void* d_ws, size_t ws_size,
                              hipStream_t stream) {
  const float* sparse = (const float*)d_in[0];
  const float* depth  = (const float*)d_in[1];
  float* out = (float*)d_out;
  (void)in_sizes; (void)n_in; (void)out_size; (void)d_ws; (void)ws_size;

  dim3 block(16, 16, 1);
  dim3 grid(WW / TILE_W, HH / TILE_H, BATCH);   // 16 x 48 x 16
  jbf_kernel<<<grid, block, 0, stream>>>(sparse, depth, out);
}